// BallQuery_19774029431168
// MI455X (gfx1250) — compile-verified
//
#include <hip/hip_runtime.h>

typedef __attribute__((ext_vector_type(2))) float v2f;
typedef __attribute__((ext_vector_type(8))) float v8f;

namespace {
constexpr int B = 4, N = 16384, M = 2048, C = 64, CT = 64, K = 32;
constexpr float RADIUS = 0.2f;
constexpr float R2 = RADIUS * RADIUS;
constexpr int MT = 16;                      // centers per tile (WMMA M)
constexpr int P1 = B * (3 + C) * M * K;     // neighbor_features elements
constexpr int P2 = B * CT * M * K;          // grouped temb elements
}

// ---------------------------------------------------------------------------
// Phase 1: ball query. One wave (32 lanes) per 16-center tile.
// Cross term via V_WMMA_F32_16X16X4_F32 with the norms folded into the
// accumulator: D = c.p - (|c|^2+|p|^2)/2 = -dist^2/2, so hit <=> D > -r^2/2.
// ---------------------------------------------------------------------------
__global__ __launch_bounds__(32) void ball_query_wmma(
    const float* __restrict__ pts, const float* __restrict__ ctr,
    int* __restrict__ idx_out) {
  const int lane = threadIdx.x;
  const int tile = blockIdx.x;
  const int b  = tile / (M / MT);
  const int m0 = (tile % (M / MT)) * MT;

  __shared__ float cn_s[MT];
  __shared__ int   idx_s[MT * K];

  const float* cb = ctr + b * 3 * M;
  const int ml = m0 + (lane & 15);
  const float cx = cb[0 * M + ml];
  const float cy = cb[1 * M + ml];
  const float cz = cb[2 * M + ml];
  if (lane < 16) cn_s[lane] = cx * cx + cy * cy + cz * cz;
  __syncthreads();

  // D vgpr r: lanes 0-15 -> center row r, lanes 16-31 -> row r+8.
  float cnv[8];
  const int rbase = (lane >> 4) << 3;
#pragma unroll
  for (int r = 0; r < 8; ++r) cnv[r] = cn_s[rbase + r];

  // A matrix 16x4 (centers x K): row m on lane m%16, K split across halves;
  // lane<16 holds K=0,1 (x,y), lane>=16 holds K=2,3 (z, zero-pad).
  v2f A;
  A.x = (lane < 16) ? cx : cz;
  A.y = (lane < 16) ? cy : 0.0f;

  const float* pb = pts + b * 3 * N;
  const int nlane = lane & 15;
  int cnt = (lane < 16) ? 0 : K;            // upper lanes report "full"

  for (int n0 = 0; n0 < N; n0 += 16) {
    const float px = pb[0 * N + n0 + nlane];
    const float py = pb[1 * N + n0 + nlane];
    const float pz = pb[2 * N + n0 + nlane];
    const float pn = px * px + py * py + pz * pz;

    // B matrix 4x16 (K x points): col n on lane n%16, K split across halves.
    v2f Bm;
    Bm.x = (lane < 16) ? px : pz;
    Bm.y = (lane < 16) ? py : 0.0f;

    v8f Cm;
#pragma unroll
    for (int r = 0; r < 8; ++r) Cm[r] = -0.5f * (cnv[r] + pn);

    v8f D = __builtin_amdgcn_wmma_f32_16x16x4_f32(
        false, A, false, Bm, (short)0, Cm, false, false);

    unsigned bal[8];
#pragma unroll
    for (int r = 0; r < 8; ++r)
      bal[r] = __builtin_amdgcn_ballot_w32(D[r] > (-0.5f * R2));

    // lane r (0..15) owns center row r; select its 16-bit hit mask
    // (ballots are wave-uniform SGPRs, selection stays in VALU).
    unsigned mask = 0;
#pragma unroll
    for (int r = 0; r < 8; ++r) {
      mask |= (lane == r)     ? (bal[r] & 0xFFFFu) : 0u;
      mask |= (lane == r + 8) ? (bal[r] >> 16)     : 0u;
    }
    if (lane < 16) {
      while (mask && cnt < K) {                 // ascending bit = scan order
        const int bit = __builtin_ctz(mask);
        mask &= mask - 1;
        idx_s[lane * K + cnt] = n0 + bit;
        ++cnt;
      }
    }
    const unsigned full = __builtin_amdgcn_ballot_w32(cnt >= K);
    if (full == 0xFFFFFFFFu) break;             // uniform early exit
  }

  // CUDA ball_query semantics: pad with first found index (0 if none).
  if (lane < 16) {
    const int first = (cnt > 0) ? idx_s[lane * K] : 0;
    for (int j = cnt; j < K; ++j) idx_s[lane * K + j] = first;
  }
  __syncthreads();

  int* ob = idx_out + (b * M + m0) * K;
  for (int t = lane; t < MT * K; t += 32) ob[t] = idx_s[t];
}

// ---------------------------------------------------------------------------
// Phase 2: gather. One thread per output element.
// d_out = [B,3+C,M,K] neighbor_features ++ [B,CT,M,K] grouped temb.
// ---------------------------------------------------------------------------
__global__ void gather_out(
    const float* __restrict__ pts, const float* __restrict__ ctr,
    const float* __restrict__ temb, const float* __restrict__ feat,
    const int* __restrict__ idx, float* __restrict__ out) {
  const int i = blockIdx.x * blockDim.x + threadIdx.x;
  if (i >= P1 + P2) return;
  if (i < P1) {
    const int k = i % K;
    const int m = (i / K) % M;
    const int c = (i / (K * M)) % (3 + C);
    const int b = i / (K * M * (3 + C));
    const int id = idx[(b * M + m) * K + k];
    float v;
    if (c < 3) {
      v = pts[(b * 3 + c) * N + id] - ctr[(b * 3 + c) * M + m];
    } else {
      v = feat[(b * C + (c - 3)) * N + id];
    }
    out[i] = v;
  } else {
    const int j = i - P1;
    const int k = j % K;
    const int m = (j / K) % M;
    const int c = (j / (K * M)) % CT;
    const int b = j / (K * M * CT);
    const int id = idx[(b * M + m) * K + k];
    out[i] = temb[(b * CT + c) * N + id];
  }
}

extern "C" void kernel_launch(void* const* d_in, const int* in_sizes, int n_in,
                              void* d_out, int out_size, void* d_ws, size_t ws_size,
                              hipStream_t stream) {
  (void)in_sizes; (void)n_in; (void)out_size; (void)ws_size;
  const float* pts  = (const float*)d_in[0];   // [B,3,N]
  const float* ctr  = (const float*)d_in[1];   // [B,3,M]
  const float* temb = (const float*)d_in[2];   // [B,CT,N]
  const float* feat = (const float*)d_in[3];   // [B,C,N]
  float* out = (float*)d_out;
  int* idx = (int*)d_ws;                       // [B,M,K] = 1 MB scratch

  ball_query_wmma<<<B * (M / MT), 32, 0, stream>>>(pts, ctr, idx);

  const int total = P1 + P2;
  gather_out<<<(total + 255) / 256, 256, 0, stream>>>(pts, ctr, temb, feat,
                                                      idx, out);
}